// Word2Vec_41695542510204
// MI455X (gfx1250) — compile-verified
//
#include <hip/hip_runtime.h>

// ---- problem constants (must match reference) ----
#define BATCH            16384
#define EMBED            128
#define NUM_NEG          4
#define NCTX             5            // 1 true context + 4 negatives
#define HASH_MASK        0x3FFFFu     // NUM_HASH_BUCKETS = 2^18
#define HSEED            1139u
#define PRIME0           2654435761u
#define PRIME1           2246822519u

// ---- LDS layout (per wave) ----
#define ROW_STRIDE       132          // 128 + 4 pad floats -> conflict-free ds_load_b64
#define WE_OFF           (NCTX * ROW_STRIDE)          // 660
#define WAVE_LDS_FLOATS  800                          // 3200 B, 16B-aligned region
#define WAVES_PER_BLOCK  8

typedef __attribute__((ext_vector_type(2))) float v2f;
typedef __attribute__((ext_vector_type(8))) float v8f;

// bucket = ((id + SEED) * prime) mod 2^18 ; exact in 32-bit since modulus | 2^32
__device__ __forceinline__ unsigned hash_bucket(unsigned id, unsigned prime) {
    return ((id + HSEED) * prime) & HASH_MASK;
}

// gather two table rows, weight-combine: returns 4 floats for e = 4*lane..4*lane+3
__device__ __forceinline__ float4 combine_rows(const float* __restrict__ table,
                                               const float* __restrict__ imp,
                                               unsigned id, int lane) {
    const unsigned b0 = hash_bucket(id, PRIME0);
    const unsigned b1 = hash_bucket(id, PRIME1);
    const float w0 = imp[(size_t)id * 2u + 0u];
    const float w1 = imp[(size_t)id * 2u + 1u];
    const float4 r0 = ((const float4*)(table + (size_t)b0 * EMBED))[lane];
    const float4 r1 = ((const float4*)(table + (size_t)b1 * EMBED))[lane];
    float4 e;
    e.x = w0 * r0.x + w1 * r1.x;
    e.y = w0 * r0.y + w1 * r1.y;
    e.z = w0 * r0.z + w1 * r1.z;
    e.w = w0 * r0.w + w1 * r1.w;
    return e;
}

__global__ __launch_bounds__(256) void Word2Vec_41695542510204_kernel(
    const int*   __restrict__ target_ids,     // [B,1]
    const int*   __restrict__ context_ids,    // [B,1]
    const int*   __restrict__ negative_ids,   // [B*4]
    const float* __restrict__ target_table,   // [2^18,128]
    const float* __restrict__ target_imp,     // [NUM_WORDS,2]
    const float* __restrict__ context_table,  // [2^18,128]
    const float* __restrict__ context_imp,    // [NUM_WORDS,2]
    float*       __restrict__ out)            // [B,5]
{
    __shared__ float lds[WAVES_PER_BLOCK * WAVE_LDS_FLOATS];

    const int lane = threadIdx.x & 31;
    const int wave = threadIdx.x >> 5;
    const int b    = blockIdx.x * WAVES_PER_BLOCK + wave;   // one batch row per wave

    float* __restrict__ wls = &lds[wave * WAVE_LDS_FLOATS];
    float* __restrict__ ceL = wls;                 // 5 rows x ROW_STRIDE
    float* __restrict__ weL = wls + WE_OFF;        // 128 floats

    // ---- stage target embedding -> LDS ----
    {
        const unsigned id = (unsigned)target_ids[b];
        const float4 we = combine_rows(target_table, target_imp, id, lane);
        ((float4*)weL)[lane] = we;                           // ds_store_b128
    }

    // ---- stage 5 context embeddings -> LDS ----
    #pragma unroll
    for (int c = 0; c < NCTX; ++c) {
        const unsigned id = (c == 0) ? (unsigned)context_ids[b]
                                     : (unsigned)negative_ids[b * NUM_NEG + (c - 1)];
        const float4 ce = combine_rows(context_table, context_imp, id, lane);
        ((float4*)(ceL + c * ROW_STRIDE))[lane] = ce;        // ds_store_b128
    }

    // make this wave's LDS writes visible before the WMMA operand reads
    asm volatile("s_wait_dscnt 0x0" ::: "memory");

    // ---- 5x128 @ 128 GEMV via V_WMMA_F32_16X16X4_F32 (exact f32) ----
    // A (16x4): lane m in [0,16): K={0,1} in v0/v1 ; lanes 16-31: K={2,3}
    // B (4x16): lanes 0-15: K={0,1} ; lanes 16-31: K={2,3} (broadcast across N)
    const int m = lane & 15;
    const int h = lane >> 4;                          // K-half select
    const int arow = (m < NCTX) ? m : 0;              // pad rows read valid data; D rows >=5 unused
    const float* __restrict__ aPtr = ceL + arow * ROW_STRIDE + 2 * h;
    const float* __restrict__ bPtr = weL + 2 * h;

    v8f acc = (v8f){0.f, 0.f, 0.f, 0.f, 0.f, 0.f, 0.f, 0.f};
    #pragma unroll
    for (int s = 0; s < EMBED / 4; ++s) {             // 32 chained K=4 steps
        const v2f a2 = *(const v2f*)(aPtr + 4 * s);   // ds_load_b64
        const v2f b2 = *(const v2f*)(bPtr + 4 * s);   // ds_load_b64 (broadcast)
        acc = __builtin_amdgcn_wmma_f32_16x16x4_f32(
                  /*neg_a=*/false, a2, /*neg_b=*/false, b2,
                  /*c_mod=*/(short)0, acc, /*reuse_a=*/false, /*reuse_b=*/false);
    }

    // D layout: VGPR j, lanes 0-15 -> M=j, N=lane. Lane 0 holds dots[0..4] in acc[0..4].
    if (lane == 0) {
        float* __restrict__ o = out + (size_t)b * NCTX;
        o[0] = acc[0];
        o[1] = acc[1];
        o[2] = acc[2];
        o[3] = acc[3];
        o[4] = acc[4];
    }
}

extern "C" void kernel_launch(void* const* d_in, const int* in_sizes, int n_in,
                              void* d_out, int out_size, void* d_ws, size_t ws_size,
                              hipStream_t stream) {
    (void)in_sizes; (void)n_in; (void)d_ws; (void)ws_size; (void)out_size;
    const int*   target_ids    = (const int*)  d_in[0];
    const int*   context_ids   = (const int*)  d_in[1];
    const int*   negative_ids  = (const int*)  d_in[2];
    const float* target_table  = (const float*)d_in[3];
    const float* target_imp    = (const float*)d_in[4];
    const float* context_table = (const float*)d_in[5];
    const float* context_imp   = (const float*)d_in[6];
    float*       out           = (float*)d_out;

    const int blocks = BATCH / WAVES_PER_BLOCK;   // 2048 blocks x 256 threads = 16384 waves
    Word2Vec_41695542510204_kernel<<<blocks, 256, 0, stream>>>(
        target_ids, context_ids, negative_ids,
        target_table, target_imp, context_table, context_imp, out);
}